// TGCNCell_29643864277042
// MI455X (gfx1250) — compile-verified
//
#include <hip/hip_runtime.h>
#include <math.h>

#define N_NODES 50000
#define IN_CH   32
#define HID     64
#define N_EDGES 800000
#define BATCH   4
#define M_ROWS  (BATCH * N_NODES)   // 200000, divisible by 16
#define C1      192                 // fused channels: [z|r|uh]

typedef __attribute__((ext_vector_type(2))) float v2f;
typedef __attribute__((ext_vector_type(8))) float v8f;

__device__ __forceinline__ float sigmf(float x) { return 1.0f / (1.0f + expf(-x)); }

// ---------------- graph normalization ----------------
__global__ void k_init_deg(float* deg) {
    int i = blockIdx.x * blockDim.x + threadIdx.x;
    if (i < N_NODES) deg[i] = 1.0f;              // self-loop weight
}

__global__ void k_scatter_deg(const long long* __restrict__ ei,
                              const float* __restrict__ w, float* deg) {
    int e = blockIdx.x * blockDim.x + threadIdx.x;
    if (e < N_EDGES) {
        int d = (int)ei[N_EDGES + e];
        atomicAdd(&deg[d], w[e]);
    }
}

__global__ void k_dinv(float* deg) {
    int i = blockIdx.x * blockDim.x + threadIdx.x;
    if (i < N_NODES) {
        float d = deg[i];
        deg[i] = (d > 0.0f) ? rsqrtf(fmaxf(d, 1e-12f)) : 0.0f;
    }
}

__global__ void k_norm(const long long* __restrict__ ei, const float* __restrict__ w,
                       const float* __restrict__ dinv, float* __restrict__ norm) {
    int e = blockIdx.x * blockDim.x + threadIdx.x;
    if (e < N_EDGES) {
        int s = (int)ei[e];
        int d = (int)ei[N_EDGES + e];
        norm[e] = dinv[s] * w[e] * dinv[d];
    }
}

// ---------------- weight concatenation ----------------
// Wx[32,192] = [W_uz | W_ur | W_uh];  Wh[64,128] = [W_hz | W_hr]
__global__ void k_concat_w(const float* Wuz, const float* Wur, const float* Wuh,
                           const float* Whz, const float* Whr,
                           float* __restrict__ Wx, float* __restrict__ Wh) {
    int i = blockIdx.x * blockDim.x + threadIdx.x;
    if (i < IN_CH * C1) {
        int k = i / C1, c = i % C1;
        float v = (c < 64) ? Wuz[k * 64 + c]
                : (c < 128) ? Wur[k * 64 + (c - 64)]
                            : Wuh[k * 64 + (c - 128)];
        Wx[i] = v;
    } else {
        int j = i - IN_CH * C1;
        if (j < HID * 128) {
            int k = j / 128, c = j % 128;
            Wh[j] = (c < 64) ? Whz[k * 64 + c] : Whr[k * 64 + (c - 64)];
        }
    }
}

// ---------------- WMMA fp32 GEMM: one wave32 per 16x16 tile, K stepped by 4 ----
// D[mt*16.., nt*16..] (+)= A[mt*16.., 0..K) @ W[0..K, nt*16..)
// A-frag (ISA 7.12.2, 32-bit A 16x4): lanes 0-15 hold K=k0,k0+1; lanes 16-31 K=k0+2,k0+3
// C/D: VGPR r -> row mt*16 + r + 8*(lane>=16), col nt*16 + lane%16
__global__ void k_wmma_gemm(const float* __restrict__ A, int lda,
                            const float* __restrict__ W, int ldw,
                            float* __restrict__ D, int ldd,
                            int K, int accum) {
    const int mt   = blockIdx.x;
    const int nt   = blockIdx.y;
    const int lane = threadIdx.x;
    const int half = lane >> 4;      // 0 or 1
    const int l16  = lane & 15;
    const int rowA  = mt * 16 + l16;
    const int col   = nt * 16 + l16;
    const int rbase = mt * 16 + half * 8;

    v8f acc = {};
    if (accum) {
#pragma unroll
        for (int r = 0; r < 8; ++r) acc[r] = D[(long long)(rbase + r) * ldd + col];
    }
    for (int k0 = 0; k0 < K; k0 += 4) {
        v2f a, b;
        a[0] = A[(long long)rowA * lda + k0 + 2 * half];
        a[1] = A[(long long)rowA * lda + k0 + 2 * half + 1];
        b[0] = W[(k0 + 2 * half) * ldw + col];
        b[1] = W[(k0 + 2 * half + 1) * ldw + col];
        acc = __builtin_amdgcn_wmma_f32_16x16x4_f32(false, a, false, b,
                                                    (short)0, acc, false, false);
    }
#pragma unroll
    for (int r = 0; r < 8; ++r) D[(long long)(rbase + r) * ldd + col] = acc[r];
}

// ---------------- aggregation: self-loop init then edge atomics ----------------
// Agg[b,n,:] = U[b,n,:] * dinv[n]^2   (self-loop term; also initializes Agg)
__global__ void k_selfloop(const float* __restrict__ U, const float* __restrict__ dinv,
                           float* __restrict__ Agg, int C) {
    long long idx = (long long)blockIdx.x * blockDim.x + threadIdx.x;
    const int CG = C >> 2;
    long long total = (long long)M_ROWS * CG;
    if (idx >= total) return;
    int c4 = (int)(idx % CG);
    long long row = idx / CG;
    int n = (int)(row % N_NODES);
    float di = dinv[n];
    float w = di * di;
    float4 u = ((const float4*)(U + row * C))[c4];
    float4 o = make_float4(u.x * w, u.y * w, u.z * w, u.w * w);
    ((float4*)(Agg + row * C))[c4] = o;
}

// Agg[b,dst,:] += U[b,src,:] * norm[e]  over all edges, all batches
__global__ void k_edge_agg(const long long* __restrict__ ei, const float* __restrict__ norm,
                           const float* __restrict__ U, float* __restrict__ Agg, int C) {
    long long idx = (long long)blockIdx.x * blockDim.x + threadIdx.x;
    const int CG = C >> 2;
    long long total = (long long)N_EDGES * BATCH * CG;
    if (idx >= total) return;
    int c4 = (int)(idx % CG);
    long long t = idx / CG;
    int b = (int)(t % BATCH);
    int e = (int)(t / BATCH);
    int s = (int)ei[e];
    int d = (int)ei[N_EDGES + e];
    float w = norm[e];
    float4 u = ((const float4*)(U + ((long long)b * N_NODES + s) * C))[c4];
    float* ap = Agg + ((long long)b * N_NODES + d) * C + c4 * 4;
    atomicAdd(ap + 0, u.x * w);
    atomicAdd(ap + 1, u.y * w);
    atomicAdd(ap + 2, u.z * w);
    atomicAdd(ap + 3, u.w * w);
}

// ---------------- GRU elementwise ----------------
// z = sig(Agg[:,0:64]+buz+bhz) -> stored back into Agg[:,0:64]
// r = sig(Agg[:,64:128]+bur+bhr); S = r*h -> stored into Agg[:,64:128]
__global__ void k_gru1(float* __restrict__ Agg, const float* __restrict__ h,
                       const float* buz, const float* bhz,
                       const float* bur, const float* bhr) {
    long long idx = (long long)blockIdx.x * blockDim.x + threadIdx.x;
    if (idx >= (long long)M_ROWS * HID) return;
    int c = (int)(idx % HID);
    long long row = idx / HID;
    float z = sigmf(Agg[row * C1 + c] + buz[c] + bhz[c]);
    float r = sigmf(Agg[row * C1 + 64 + c] + bur[c] + bhr[c]);
    Agg[row * C1 + c]      = z;
    Agg[row * C1 + 64 + c] = r * h[row * HID + c];
}

// h_hat = tanh(Agg[:,128:192]+buh + Agg2+bhh); out = (1-z)*h + z*h_hat
__global__ void k_gru2(const float* __restrict__ Agg, const float* __restrict__ Agg2,
                       const float* __restrict__ h, const float* buh, const float* bhh,
                       float* __restrict__ out) {
    long long idx = (long long)blockIdx.x * blockDim.x + threadIdx.x;
    if (idx >= (long long)M_ROWS * HID) return;
    int c = (int)(idx % HID);
    long long row = idx / HID;
    float hhat = tanhf(Agg[row * C1 + 128 + c] + buh[c] + Agg2[row * HID + c] + bhh[c]);
    float z  = Agg[row * C1 + c];
    float hv = h[row * HID + c];
    out[idx] = (1.0f - z) * hv + z * hhat;
}

extern "C" void kernel_launch(void* const* d_in, const int* in_sizes, int n_in,
                              void* d_out, int out_size, void* d_ws, size_t ws_size,
                              hipStream_t stream) {
    const float*      x   = (const float*)d_in[0];
    const float*      h   = (const float*)d_in[1];
    const long long*  ei  = (const long long*)d_in[2];
    const float*      ew  = (const float*)d_in[3];
    const float* Wuz = (const float*)d_in[4];  const float* buz = (const float*)d_in[5];
    const float* Whz = (const float*)d_in[6];  const float* bhz = (const float*)d_in[7];
    const float* Wur = (const float*)d_in[8];  const float* bur = (const float*)d_in[9];
    const float* Whr = (const float*)d_in[10]; const float* bhr = (const float*)d_in[11];
    const float* Wuh = (const float*)d_in[12]; const float* buh = (const float*)d_in[13];
    const float* Whh = (const float*)d_in[14]; const float* bhh = (const float*)d_in[15];
    float* out = (float*)d_out;
    float* ws  = (float*)d_ws;

    // workspace layout (floats, 16B-aligned slices)
    float* dinv = ws;                                   // 50000  (deg -> dinv in place)
    float* norm = ws + 50176;                           // 800000
    float* Wx   = ws + 850176;                          // 32*192
    float* Wh   = ws + 856320;                          // 64*128
    float* U    = ws + 864512;                          // M*192
    float* Agg  = U + (long long)M_ROWS * C1;           // M*192
    float* T    = U;                                    // reuse: M*64
    float* Agg2 = U + (long long)M_ROWS * HID;          // reuse: M*64

    // 1) symmetric GCN normalization
    k_init_deg   <<<(N_NODES + 255) / 256, 256, 0, stream>>>(dinv);
    k_scatter_deg<<<(N_EDGES + 255) / 256, 256, 0, stream>>>(ei, ew, dinv);
    k_dinv       <<<(N_NODES + 255) / 256, 256, 0, stream>>>(dinv);
    k_norm       <<<(N_EDGES + 255) / 256, 256, 0, stream>>>(ei, ew, dinv, norm);

    // 2) fused weights
    k_concat_w<<<(IN_CH * C1 + HID * 128 + 255) / 256, 256, 0, stream>>>(
        Wuz, Wur, Wuh, Whz, Whr, Wx, Wh);

    // 3) U = x@Wx ; U[:, :128] += h@Wh   (WMMA fp32)
    k_wmma_gemm<<<dim3(M_ROWS / 16, C1 / 16), 32, 0, stream>>>(x, IN_CH, Wx, C1, U, C1, IN_CH, 0);
    k_wmma_gemm<<<dim3(M_ROWS / 16, 128 / 16), 32, 0, stream>>>(h, HID, Wh, 128, U, C1, HID, 1);

    // 4) propagation pass 1 over 192 fused channels
    long long t1 = (long long)M_ROWS * (C1 / 4);
    k_selfloop<<<(unsigned)((t1 + 255) / 256), 256, 0, stream>>>(U, dinv, Agg, C1);
    long long t2 = (long long)N_EDGES * BATCH * (C1 / 4);
    k_edge_agg<<<(unsigned)((t2 + 255) / 256), 256, 0, stream>>>(ei, norm, U, Agg, C1);

    // 5) gates: z, S=r*h (in place in Agg)
    long long t3 = (long long)M_ROWS * HID;
    k_gru1<<<(unsigned)((t3 + 255) / 256), 256, 0, stream>>>(Agg, h, buz, bhz, bur, bhr);

    // 6) T = S @ W_hh   (S lives at Agg[:,64:128], lda=192)
    k_wmma_gemm<<<dim3(M_ROWS / 16, HID / 16), 32, 0, stream>>>(Agg + 64, C1, Whh, HID, T, HID, HID, 0);

    // 7) propagation pass 2 over 64 channels
    long long t4 = (long long)M_ROWS * (HID / 4);
    k_selfloop<<<(unsigned)((t4 + 255) / 256), 256, 0, stream>>>(T, dinv, Agg2, HID);
    long long t5 = (long long)N_EDGES * BATCH * (HID / 4);
    k_edge_agg<<<(unsigned)((t5 + 255) / 256), 256, 0, stream>>>(ei, norm, T, Agg2, HID);

    // 8) h_hat + final blend
    k_gru2<<<(unsigned)((t3 + 255) / 256), 256, 0, stream>>>(Agg, Agg2, h, buh, bhh, out);
}